// LSEPLoss_1580547965396
// MI455X (gfx1250) — compile-verified
//
#include <hip/hip_runtime.h>

// ---------------------------------------------------------------------------
// LSEP loss, factorized form:
//   lsep_i = M + log(exp(-M) + S0*S1),  M = pmax - pmin
//   S0 = sum_{t=0} exp(p - pmax),  S1 = sum_{t=1} exp(pmin - p)
//   out = mean_i lsep_i
// O(n*C) instead of the reference's O(n*C^2) pair tensor.
// ---------------------------------------------------------------------------

#define C_DIM 1024
#define BLOCK 256
#define VPT (C_DIM / BLOCK)  // 4 elements per thread
#define NWAVES (BLOCK / 32)  // 8 wave32 waves per workgroup

typedef __attribute__((ext_vector_type(2))) float v2f;
typedef __attribute__((ext_vector_type(4))) float v4f;
typedef __attribute__((ext_vector_type(8))) float v8f;
typedef int v4i __attribute__((vector_size(16)));  // matches builtin's V4i

#if __has_builtin(__builtin_amdgcn_global_load_async_to_lds_b128) && \
    __has_builtin(__builtin_amdgcn_s_wait_asynccnt)
#define HAVE_ASYNC_LDS 1
typedef __attribute__((address_space(1))) v4i* gas_v4i_ptr;
typedef __attribute__((address_space(3))) v4i* lds_v4i_ptr;
#else
#define HAVE_ASYNC_LDS 0
#endif

__device__ __forceinline__ float wave_max(float v) {
#pragma unroll
  for (int o = 16; o > 0; o >>= 1) v = fmaxf(v, __shfl_xor(v, o, 32));
  return v;
}
__device__ __forceinline__ float wave_min(float v) {
#pragma unroll
  for (int o = 16; o > 0; o >>= 1) v = fminf(v, __shfl_xor(v, o, 32));
  return v;
}

// Sum all 32 lane values of `a` using V_WMMA_F32_16X16X4_F32 with B = ones.
// A layout (16x4 f32): VGPR0 lanes0-15 -> (M=lane,K=0), lanes16-31 -> (M=lane-16,K=2).
// With A = (a, 0) and B = all-ones, D[m][n] = a[m] + a[m+16] for every n.
// Per-lane sum of the 8 D VGPRs gives the sum over that lane-half's 8 M rows;
// one shfl_xor(16) combines the halves -> every lane holds the full 32-lane sum.
__device__ __forceinline__ float wmma_lane_sum(float a) {
  v2f A;
  A.x = a;
  A.y = 0.0f;
  v2f B;
  B.x = 1.0f;
  B.y = 1.0f;
  v8f Z = {};
  v8f D = __builtin_amdgcn_wmma_f32_16x16x4_f32(
      /*neg_a=*/false, A, /*neg_b=*/false, B,
      /*c_mod=*/(short)0, Z, /*reuse_a=*/false, /*reuse_b=*/false);
  float t = 0.0f;
#pragma unroll
  for (int r = 0; r < 8; ++r) t += D[r];
  t += __shfl_xor(t, 16, 32);
  return t;
}

__global__ void __launch_bounds__(BLOCK)
lsep_row_kernel(const float* __restrict__ preds,
                const float* __restrict__ targets,
                float* __restrict__ row_out) {
  __shared__ alignas(16) float s_p[C_DIM];
  __shared__ alignas(16) float s_t[C_DIM];
  __shared__ float s_r0[BLOCK];
  __shared__ float s_r1[BLOCK];
  __shared__ float s_mx[NWAVES];
  __shared__ float s_mn[NWAVES];

  const int tid = threadIdx.x;
  const size_t base = (size_t)blockIdx.x * C_DIM + (size_t)tid * VPT;

  float p[VPT], t[VPT];
#if HAVE_ASYNC_LDS
  // CDNA5 async global->LDS staging (ASYNCcnt path), 16B per lane, coalesced.
  __builtin_amdgcn_global_load_async_to_lds_b128(
      (gas_v4i_ptr)(preds + base), (lds_v4i_ptr)(s_p + tid * VPT), 0, 0);
  __builtin_amdgcn_global_load_async_to_lds_b128(
      (gas_v4i_ptr)(targets + base), (lds_v4i_ptr)(s_t + tid * VPT), 0, 0);
  __builtin_amdgcn_s_wait_asynccnt(0);
  // Each thread only reads the LDS region its own async loads wrote.
#pragma unroll
  for (int k = 0; k < VPT; ++k) {
    p[k] = s_p[tid * VPT + k];
    t[k] = s_t[tid * VPT + k];
  }
#else
  {
    v4f pv = *(const v4f*)(preds + base);
    v4f tv = *(const v4f*)(targets + base);
#pragma unroll
    for (int k = 0; k < VPT; ++k) {
      p[k] = pv[k];
      t[k] = tv[k];
    }
  }
#endif

  // ---- pass 1: row max / min ----
  float mx = p[0], mn = p[0];
#pragma unroll
  for (int k = 1; k < VPT; ++k) {
    mx = fmaxf(mx, p[k]);
    mn = fminf(mn, p[k]);
  }
  mx = wave_max(mx);
  mn = wave_min(mn);
  if ((tid & 31) == 0) {
    s_mx[tid >> 5] = mx;
    s_mn[tid >> 5] = mn;
  }
  __syncthreads();
  float pmax = s_mx[0], pmin = s_mn[0];
#pragma unroll
  for (int w = 1; w < NWAVES; ++w) {
    pmax = fmaxf(pmax, s_mx[w]);
    pmin = fminf(pmin, s_mn[w]);
  }

  // ---- pass 2: factorized masked-exp sums ----
  float s0 = 0.0f, s1 = 0.0f;
#pragma unroll
  for (int k = 0; k < VPT; ++k) {
    if (t[k] < 0.5f)
      s0 += __expf(p[k] - pmax);  // arguments <= 0, no overflow
    else
      s1 += __expf(pmin - p[k]);
  }
  s_r0[tid] = s0;
  s_r1[tid] = s1;
  __syncthreads();

  // ---- final reduction on wave 0: serial fold 256->32, WMMA fold 32->1 ----
  if (tid < 32) {  // whole wave 0 active => EXEC all ones for WMMA
    float a0 = 0.0f, a1 = 0.0f;
#pragma unroll
    for (int j = 0; j < NWAVES; ++j) {
      a0 += s_r0[tid + 32 * j];
      a1 += s_r1[tid + 32 * j];
    }
    float S0 = wmma_lane_sum(a0);
    float S1 = wmma_lane_sum(a1);
    if (tid == 0) {
      float M = pmax - pmin;  // >= 0
      row_out[blockIdx.x] = M + logf(__expf(-M) + S0 * S1);
    }
  }
}

// Deterministic mean over the 128 per-row results (no float atomics).
__global__ void __launch_bounds__(128)
lsep_mean_kernel(const float* __restrict__ row_in, float* __restrict__ out, int n) {
  __shared__ float s_w[4];
  const int tid = threadIdx.x;
  float v = (tid < n) ? row_in[tid] : 0.0f;
#pragma unroll
  for (int o = 16; o > 0; o >>= 1) v += __shfl_xor(v, o, 32);
  if ((tid & 31) == 0) s_w[tid >> 5] = v;
  __syncthreads();
  if (tid == 0) {
    float s = s_w[0] + s_w[1] + s_w[2] + s_w[3];
    out[0] = s / (float)n;
  }
}

extern "C" void kernel_launch(void* const* d_in, const int* in_sizes, int n_in,
                              void* d_out, int out_size, void* d_ws, size_t ws_size,
                              hipStream_t stream) {
  (void)n_in;
  (void)out_size;
  (void)ws_size;
  const float* preds = (const float*)d_in[0];
  const float* targets = (const float*)d_in[1];
  float* out = (float*)d_out;
  float* rows = (float*)d_ws;  // n floats of scratch

  const int n = in_sizes[0] / C_DIM;  // 128 rows

  lsep_row_kernel<<<n, BLOCK, 0, stream>>>(preds, targets, rows);
  lsep_mean_kernel<<<1, 128, 0, stream>>>(rows, out, n);
}